// BaseLayerGate_76493367542579
// MI455X (gfx1250) — compile-verified
//
#include <hip/hip_runtime.h>
#include <hip/hip_bf16.h>

// ---------------------------------------------------------------------------
// BaseLayer balanced-gating for MI455X (gfx1250, wave32)
//   Phase 1: affinities = features @ W^T + b   via V_WMMA_F32_16X16X4_F32
//   Phase 2: sequential greedy balanced assignment + softmax gather
// ---------------------------------------------------------------------------

#define T_TOKENS   8192
#define D_MODEL    2048
#define N_EXPERTS  16
#define CAP        (T_TOKENS / N_EXPERTS)   // 512

#define BK         64                        // K-chunk staged in LDS
#define ROWS       64                        // tokens per block (4 waves x 16)
#define LDS_STRIDE 68                        // 64 + 4 pad -> conflict-free ds_load_b64

typedef float v2f __attribute__((ext_vector_type(2)));
typedef float v4f __attribute__((ext_vector_type(4)));
typedef float v8f __attribute__((ext_vector_type(8)));

// ---------------------------------------------------------------------------
// Phase 1: 128 blocks x 128 threads (4 waves). Each wave owns a 16x16 tile of
// affinities (16 tokens x 16 experts). fp32 WMMA, K-loop of 2048 in chunks of
// 64 staged through LDS with coalesced b128 loads.
// ---------------------------------------------------------------------------
__global__ __launch_bounds__(128)
void gate_gemm_wmma(const float* __restrict__ feat,
                    const float* __restrict__ W,
                    const float* __restrict__ bias,
                    float* __restrict__ affin) {
    __shared__ float ldsA[ROWS * LDS_STRIDE];       // 64 x 68 floats
    __shared__ float ldsB[N_EXPERTS * LDS_STRIDE];  // 16 x 68 floats

    const int tid  = threadIdx.x;
    const int wave = tid >> 5;          // 0..3
    const int lane = tid & 31;
    const int rowBase = blockIdx.x * ROWS;
    const int waveRow = rowBase + wave * 16;

    // WMMA 32-bit operand lane mapping (ISA 7.12.2):
    //   A 16x4 : lane<16 -> M=lane, K={k,k+1}; lane>=16 -> M=lane-16, K={k+2,k+3}
    //   B 4x16 : lane<16 -> N=lane, K={k,k+1}; lane>=16 -> N=lane-16, K={k+2,k+3}
    const int mrow = lane & 15;
    const int koff = (lane >> 4) * 2;

    v8f acc = {};

    for (int k0 = 0; k0 < D_MODEL; k0 += BK) {
        // ---- stage A tile: 64 rows x 64 cols = 1024 float4 slots, 128 thr ----
        #pragma unroll
        for (int i = 0; i < 8; ++i) {
            int s  = tid + i * 128;
            int r  = s >> 4;            // row 0..63
            int c4 = s & 15;            // float4 column 0..15
            v4f v = *(const v4f*)(feat + (size_t)(rowBase + r) * D_MODEL + k0 + c4 * 4);
            *(v4f*)(&ldsA[r * LDS_STRIDE + c4 * 4]) = v;
        }
        // ---- stage B tile (W chunk): 16 x 64 = 256 float4 slots ----
        #pragma unroll
        for (int i = 0; i < 2; ++i) {
            int s  = tid + i * 128;
            int r  = s >> 4;            // expert 0..15
            int c4 = s & 15;
            v4f v = *(const v4f*)(W + (size_t)r * D_MODEL + k0 + c4 * 4);
            *(v4f*)(&ldsB[r * LDS_STRIDE + c4 * 4]) = v;
        }
        // prefetch next A chunk into L2 while we compute (global_prefetch_b8)
        if (k0 + BK < D_MODEL)
            __builtin_prefetch(feat + (size_t)(rowBase + (tid >> 1)) * D_MODEL
                                    + k0 + BK + (tid & 1) * 32, 0, 1);
        __syncthreads();

        const float* aPtr = &ldsA[(wave * 16 + mrow) * LDS_STRIDE + koff];
        const float* bPtr = &ldsB[mrow * LDS_STRIDE + koff];
        #pragma unroll
        for (int kk = 0; kk < BK; kk += 4) {
            v2f a = *(const v2f*)(aPtr + kk);
            v2f b = *(const v2f*)(bPtr + kk);
#if defined(__has_builtin) && __has_builtin(__builtin_amdgcn_wmma_f32_16x16x4_f32)
            acc = __builtin_amdgcn_wmma_f32_16x16x4_f32(
                /*neg_a=*/false, a, /*neg_b=*/false, b,
                /*c_mod=*/(short)0, acc, /*reuse_a=*/false, /*reuse_b=*/false);
#else
            // scalar fallback reproducing the same C lane layout
            {
                int n  = lane & 15;
                int mB = (lane >> 4) * 8;
                #pragma unroll
                for (int i = 0; i < 8; ++i) {
                    float s0 = acc[i];
                    #pragma unroll
                    for (int kq = 0; kq < 4; ++kq)
                        s0 = fmaf(ldsA[(wave * 16 + mB + i) * LDS_STRIDE + kk + kq],
                                  ldsB[n * LDS_STRIDE + kk + kq], s0);
                    acc[i] = s0;
                }
            }
#endif
        }
        __syncthreads();
    }

    // ---- epilogue: C/D layout VGPR i -> (M = i + 8*(lane>=16), N = lane&15) ----
    const int n  = lane & 15;
    const int mB = (lane >> 4) * 8;
    const float bb = bias[n];
    #pragma unroll
    for (int i = 0; i < 8; ++i) {
        affin[(size_t)(waveRow + mB + i) * N_EXPERTS + n] = acc[i] + bb;
    }
}

// ---------------------------------------------------------------------------
// Phase 2: single-wave sequential greedy balanced assignment (matches the
// reference lax.scan semantics exactly, incl. argmax first-max tie-break).
// Lanes 0..15 = experts; lanes 16..31 mirror so EXEC stays full and shuffles
// (xor 1,2,4,8) remain within each 16-lane group.
// Outputs (all float): out[0..T)   = sort_by_expert (token ids grouped by expert)
//                      out[T..2T)  = gathered softmax score per token
// ---------------------------------------------------------------------------
#define BATCH 256

__global__ __launch_bounds__(32)
void gate_assign(const float* __restrict__ affin, float* __restrict__ out) {
    __shared__ float rows[BATCH * N_EXPERTS];   // 16 KB batch of affinity rows

    const int lane = threadIdx.x;
    const int e    = lane & 15;
    int cap = CAP;      // remaining capacity of expert `e` (mirrored upper half)
    int cnt = 0;        // tokens already assigned to expert `e`

    for (int t0 = 0; t0 < T_TOKENS; t0 += BATCH) {
        // cooperative coalesced batch load: 256 tokens x 16 floats = 1024 float4
        #pragma unroll
        for (int i = 0; i < 32; ++i) {
            int s = lane + i * 32;
            *(v4f*)(&rows[s * 4]) =
                *(const v4f*)(affin + (size_t)t0 * N_EXPERTS + s * 4);
        }
        __syncthreads();

        for (int tl = 0; tl < BATCH; ++tl) {
            float a = rows[tl * N_EXPERTS + e];

            // softmax pieces (off the cap-dependency critical path)
            float m = a;
            #pragma unroll
            for (int off = 8; off >= 1; off >>= 1)
                m = fmaxf(m, __shfl_xor(m, off, 32));
            float ex = __expf(a - m);
            float sum = ex;
            #pragma unroll
            for (int off = 8; off >= 1; off >>= 1)
                sum += __shfl_xor(sum, off, 32);

            // capacity-masked argmax, tie-break toward smaller expert index
            float av = (cap > 0) ? a : -__builtin_inff();
            int   ai = e;
            #pragma unroll
            for (int off = 8; off >= 1; off >>= 1) {
                float ov = __shfl_xor(av, off, 32);
                int   oi = __shfl_xor(ai, off, 32);
                if (ov > av || (ov == av && oi < ai)) { av = ov; ai = oi; }
            }

            if (e == ai) {                       // winner lane (in both halves)
                if (lane < 16) {
                    out[ai * CAP + cnt]       = (float)(t0 + tl);  // sort_by_expert
                    out[T_TOKENS + t0 + tl]   = ex / sum;          // gathered score
                }
                ++cnt;
                --cap;
            }
        }
        __syncthreads();
    }
}

// ---------------------------------------------------------------------------
extern "C" void kernel_launch(void* const* d_in, const int* in_sizes, int n_in,
                              void* d_out, int out_size, void* d_ws, size_t ws_size,
                              hipStream_t stream) {
    const float* feat = (const float*)d_in[0];   // [8192 x 2048]
    const float* W    = (const float*)d_in[1];   // [16 x 2048]
    const float* bias = (const float*)d_in[2];   // [16]
    // d_in[3] = is_training (unused)

    float* affin = (float*)d_ws;                 // [8192 x 16] fp32 scratch
    float* out   = (float*)d_out;                // [8192] sort + [8192] scores

    gate_gemm_wmma<<<dim3(T_TOKENS / ROWS), dim3(128), 0, stream>>>(feat, W, bias, affin);
    gate_assign  <<<dim3(1),               dim3(32),  0, stream>>>(affin, out);
}